// ScoreModelHead_81784767250905
// MI455X (gfx1250) — compile-verified
//
#include <hip/hip_runtime.h>
#include <hip/hip_bf16.h>
#include <math.h>

// ---------------------------------------------------------------------------
// Problem constants
// ---------------------------------------------------------------------------
#define N_T    32
#define N_Q    256
#define N_PTS  (N_T * N_Q)      // 8192 query points
#define N_NODE 8192
#define D_FEAT 128
#define CH     32
#define K_NN   16
#define NB_RBF 8
#define GAMMA  4.0f
#define INV_ANG_MULT 0.70710678118654752440f  // 1/sqrt(2)

typedef __attribute__((ext_vector_type(2))) float v2f;
typedef __attribute__((ext_vector_type(8))) float v8f;

// ---------------------------------------------------------------------------
// quaternion rotate: out = v + 2*(w*(u x v) + u x (u x v))
// ---------------------------------------------------------------------------
__device__ __forceinline__ void quat_rot(float w, float x, float y, float z,
                                         float vx, float vy, float vz,
                                         float& ox, float& oy, float& oz) {
    float uvx = y * vz - z * vy;
    float uvy = z * vx - x * vz;
    float uvz = x * vy - y * vx;
    float uuvx = y * uvz - z * uvy;
    float uuvy = z * uvx - x * uvz;
    float uuvz = x * uvy - y * uvx;
    ox = vx + 2.0f * (w * uvx + uuvx);
    oy = vy + 2.0f * (w * uvy + uuvy);
    oz = vz + 2.0f * (w * uvz + uuvz);
}

// ---------------------------------------------------------------------------
// Kernel 0: tiny precompute.
//   tw[j]   = 1 + (time_emb @ W_time)[j]
//   coef[]  = collapsed dtp weights: a/b/g for lin then ang (6 x 32)
// ---------------------------------------------------------------------------
__global__ void precompute_kernel(const float* __restrict__ time_emb,
                                  const float* __restrict__ W_time,
                                  const float* __restrict__ w_path_lin,
                                  const float* __restrict__ W_vec_lin,
                                  const float* __restrict__ w_path_ang,
                                  const float* __restrict__ W_vec_ang,
                                  float* __restrict__ tw,
                                  float* __restrict__ coef) {
    int tid = threadIdx.x;  // 128 threads
    float acc = 0.0f;
    for (int i = 0; i < D_FEAT; ++i) acc += time_emb[i] * W_time[i * D_FEAT + tid];
    tw[tid] = 1.0f + acc;

    if (tid < CH) {
        int c = tid;
        float m_su = 0.f, m_vt = 0.f, m_x = 0.f;
        float a_su = 0.f, a_vt = 0.f, a_x = 0.f;
        for (int o = 0; o < CH; ++o) {
            m_su += W_vec_lin[(c)      * CH + o];
            m_vt += W_vec_lin[(CH + c) * CH + o];
            m_x  += W_vec_lin[(2*CH+c) * CH + o];
            a_su += W_vec_ang[(c)      * CH + o];
            a_vt += W_vec_ang[(CH + c) * CH + o];
            a_x  += W_vec_ang[(2*CH+c) * CH + o];
        }
        const float inv = 1.0f / (float)CH;  // mean over output channels
        coef[0*CH + c] = w_path_lin[2*CH + c] * m_su * inv;
        coef[1*CH + c] = w_path_lin[3*CH + c] * m_vt * inv;
        coef[2*CH + c] = w_path_lin[4*CH + c] * m_x  * inv;
        coef[3*CH + c] = w_path_ang[2*CH + c] * a_su * inv;
        coef[4*CH + c] = w_path_ang[3*CH + c] * a_vt * inv;
        coef[5*CH + c] = w_path_ang[4*CH + c] * a_x  * inv;
    }
}

// ---------------------------------------------------------------------------
// Kernel 1: NF = node_feature @ W_ext  (8192x128 @ 128x128, f32)
// One wave per 16x16 output tile; V_WMMA_F32_16X16X4_F32, 32 K-steps.
// A frag (16x4 f32): lane l holds M=l&15, K-pair 2*(l>>4)+{0,1}.
// B frag (4x16 f32): lane l holds N=l&15, K-pair 2*(l>>4)+{0,1}.
// C/D (16x16 f32): VGPR r holds M = r + 8*(l>>4), N = l&15.
// ---------------------------------------------------------------------------
__global__ void __launch_bounds__(256)
gemm_nf_kernel(const float* __restrict__ A,      // node_feature [8192,128]
               const float* __restrict__ Bw,     // W_ext        [128,128]
               float* __restrict__ C) {          // NF           [8192,128]
    const int lane = threadIdx.x & 31;
    const int wave = threadIdx.x >> 5;
    const int tile = blockIdx.x * 8 + wave;      // 4096 tiles total
    const int tm = tile >> 3;                    // 0..511 (M tiles)
    const int tn = tile & 7;                     // 0..7   (N tiles)
    const int m    = lane & 15;
    const int half = lane >> 4;

    v8f acc = {};
    const float* arow = A + (tm * 16 + m) * D_FEAT;
    const int ncol = tn * 16 + m;

    for (int kk = 0; kk < D_FEAT; kk += 4) {
        const int k2 = kk + 2 * half;
        v2f a, b;
        a.x = arow[k2];
        a.y = arow[k2 + 1];
        b.x = Bw[(k2)     * D_FEAT + ncol];
        b.y = Bw[(k2 + 1) * D_FEAT + ncol];
        acc = __builtin_amdgcn_wmma_f32_16x16x4_f32(
            /*neg_a=*/false, a, /*neg_b=*/false, b,
            /*c_mod=*/(short)0, acc, /*reuse_a=*/false, /*reuse_b=*/false);
    }

#pragma unroll
    for (int r = 0; r < 8; ++r) {
        const int row = tm * 16 + r + 8 * half;
        C[row * D_FEAT + ncol] = acc[r];
    }
}

// ---------------------------------------------------------------------------
// Kernel 2: KNN. One thread per query point n = t*256 + q.
// Node coords/batch staged through LDS in 2048-node (32KB) chunks.
// Register top-16 via guarded ripple insert (ties keep lower index,
// matching jax.lax.top_k).
// ---------------------------------------------------------------------------
__global__ void __launch_bounds__(256)
knn_kernel(const float* __restrict__ T,           // [32,7]
           const float* __restrict__ query_coord, // [256,3]
           const int*   __restrict__ query_batch, // [256]
           const float* __restrict__ node_coord,  // [8192,3]
           const int*   __restrict__ node_batch,  // [8192]
           int*   __restrict__ knn_idx,           // [8192,16]
           float* __restrict__ knn_d) {           // [8192,16]
    __shared__ float s_nx[2048], s_ny[2048], s_nz[2048];
    __shared__ int   s_nb[2048];

    const int n = blockIdx.x * blockDim.x + threadIdx.x;  // 0..8191
    const int t = n >> 8;
    const int q = n & 255;

    const float qw0 = T[t*7+0], qx = T[t*7+1], qy = T[t*7+2], qz = T[t*7+3];
    const float X0  = T[t*7+4], X1 = T[t*7+5], X2 = T[t*7+6];
    const float cx = query_coord[q*3+0], cy = query_coord[q*3+1], cz = query_coord[q*3+2];
    float px, py, pz;
    quat_rot(qw0, qx, qy, qz, cx, cy, cz, px, py, pz);
    px += X0; py += X1; pz += X2;
    const int myb = query_batch[q];

    float bd[K_NN];
    int   bi[K_NN];
#pragma unroll
    for (int k = 0; k < K_NN; ++k) { bd[k] = 3.0e38f; bi[k] = 0; }

    for (int base = 0; base < N_NODE; base += 2048) {
        for (int j = threadIdx.x; j < 2048; j += blockDim.x) {
            s_nx[j] = node_coord[(base + j) * 3 + 0];
            s_ny[j] = node_coord[(base + j) * 3 + 1];
            s_nz[j] = node_coord[(base + j) * 3 + 2];
            s_nb[j] = node_batch[base + j];
        }
        __syncthreads();
        for (int j = 0; j < 2048; ++j) {
            const float dx = s_nx[j] - px;
            const float dy = s_ny[j] - py;
            const float dz = s_nz[j] - pz;
            float d2 = dx*dx + dy*dy + dz*dz;
            d2 = (s_nb[j] == myb) ? d2 : 1.0e9f;
            if (d2 < bd[K_NN - 1]) {
                float dd = d2; int ii = base + j;
#pragma unroll
                for (int k = 0; k < K_NN; ++k) {
                    const bool sm = dd < bd[k];
                    const float td = sm ? bd[k] : dd;
                    const int   ti = sm ? bi[k] : ii;
                    bd[k] = sm ? dd : bd[k];
                    bi[k] = sm ? ii : bi[k];
                    dd = td; ii = ti;
                }
            }
        }
        __syncthreads();
    }

#pragma unroll
    for (int k = 0; k < K_NN; ++k) {
        knn_idx[n * K_NN + k] = bi[k];
        knn_d  [n * K_NN + k] = sqrtf(fmaxf(bd[k], 1e-12f));
    }
}

// ---------------------------------------------------------------------------
// Kernel 3: field extraction + collapsed dtp + per-block partial reduction.
// Grid: 1024 blocks (32 per t), 256 threads (8 waves), one wave per query.
// Phase A: field[128] into LDS (lane handles 4 columns).
// Phase B: lane = channel c; dtp vec contributions; wave shuffle-reduce;
//          lane 0 rotates by q^-1, orbital cross, query-weight scale.
// Deterministic: fixed-shape tree reductions, no atomics.
// ---------------------------------------------------------------------------
__global__ void __launch_bounds__(256)
field_dtp_kernel(const float* __restrict__ T,
                 const float* __restrict__ query_weight,
                 const float* __restrict__ query_feature,
                 const float* __restrict__ query_coord,
                 const float* __restrict__ W_rbf,     // [8,128]
                 const float* __restrict__ NF,        // [8192,128]
                 const float* __restrict__ tw,        // [128]
                 const float* __restrict__ coef,      // [6,32]
                 const int*   __restrict__ knn_idx,
                 const float* __restrict__ knn_d,
                 float* __restrict__ partial) {       // [1024,6]
    __shared__ float fld[8][D_FEAT];
    __shared__ float red[8][6];

    const int wave = threadIdx.x >> 5;
    const int lane = threadIdx.x & 31;
    const int t = blockIdx.x >> 5;                      // 0..31
    const int q = ((blockIdx.x & 31) << 3) + wave;      // 0..255
    const int n = t * N_Q + q;

    // ---- Phase A: field ----
    const int j0 = lane * 4;
    float wr[NB_RBF][4];
#pragma unroll
    for (int b = 0; b < NB_RBF; ++b)
#pragma unroll
        for (int c = 0; c < 4; ++c)
            wr[b][c] = W_rbf[b * D_FEAT + j0 + c];

    int   nidx[K_NN];
    float nd[K_NN];
#pragma unroll
    for (int k = 0; k < K_NN; ++k) {
        nidx[k] = knn_idx[n * K_NN + k];
        nd[k]   = knn_d  [n * K_NN + k];
        __builtin_prefetch(NF + (size_t)nidx[k] * D_FEAT + j0, 0, 1);
    }

    float acc0 = 0.f, acc1 = 0.f, acc2 = 0.f, acc3 = 0.f;
#pragma unroll
    for (int k = 0; k < K_NN; ++k) {
        const float d = nd[k];
        float w0 = 0.f, w1 = 0.f, w2 = 0.f, w3 = 0.f;
#pragma unroll
        for (int b = 0; b < NB_RBF; ++b) {
            const float dd = d - (3.0f / 7.0f) * (float)b;  // linspace(0,3,8)
            const float rb = __expf(-GAMMA * dd * dd);
            w0 += rb * wr[b][0];
            w1 += rb * wr[b][1];
            w2 += rb * wr[b][2];
            w3 += rb * wr[b][3];
        }
        const float4 nf = *(const float4*)(NF + (size_t)nidx[k] * D_FEAT + j0);
        acc0 += w0 * nf.x;
        acc1 += w1 * nf.y;
        acc2 += w2 * nf.z;
        acc3 += w3 * nf.w;
    }
    fld[wave][j0 + 0] = acc0 * tw[j0 + 0];
    fld[wave][j0 + 1] = acc1 * tw[j0 + 1];
    fld[wave][j0 + 2] = acc2 * tw[j0 + 2];
    fld[wave][j0 + 3] = acc3 * tw[j0 + 3];
    __syncthreads();

    // ---- Phase B: dtp (collapsed) ----
    const int c = lane;
    const float tc = fld[wave][c];
    const float ux = fld[wave][CH + 3*c + 0];
    const float uy = fld[wave][CH + 3*c + 1];
    const float uz = fld[wave][CH + 3*c + 2];

    const float sc  = query_feature[q * D_FEAT + c];
    const float w0x = query_feature[q * D_FEAT + CH + 3*c + 0];
    const float w0y = query_feature[q * D_FEAT + CH + 3*c + 1];
    const float w0z = query_feature[q * D_FEAT + CH + 3*c + 2];

    const float qw0 = T[t*7+0], qx = T[t*7+1], qy = T[t*7+2], qz = T[t*7+3];
    float vx, vy, vz;
    quat_rot(qw0, qx, qy, qz, w0x, w0y, w0z, vx, vy, vz);

    const float cxv = vy * uz - vz * uy;   // v x u
    const float cyv = vz * ux - vx * uz;
    const float czv = vx * uy - vy * ux;

    const float al = coef[0*CH + c], bl = coef[1*CH + c], gl = coef[2*CH + c];
    const float aa = coef[3*CH + c], ba = coef[4*CH + c], ga = coef[5*CH + c];

    float lx = al * sc * ux + bl * vx * tc + gl * cxv;
    float ly = al * sc * uy + bl * vy * tc + gl * cyv;
    float lz = al * sc * uz + bl * vz * tc + gl * czv;
    float ax = aa * sc * ux + ba * vx * tc + ga * cxv;
    float ay = aa * sc * uy + ba * vy * tc + ga * cyv;
    float az = aa * sc * uz + ba * vz * tc + ga * czv;

    // wave32 shuffle tree reduction over channels
#pragma unroll
    for (int off = 16; off >= 1; off >>= 1) {
        lx += __shfl_xor(lx, off, 32);
        ly += __shfl_xor(ly, off, 32);
        lz += __shfl_xor(lz, off, 32);
        ax += __shfl_xor(ax, off, 32);
        ay += __shfl_xor(ay, off, 32);
        az += __shfl_xor(az, off, 32);
    }

    if (lane == 0) {
        const float inv = 1.0f / (float)CH;   // mean over channels
        float lvx = lx * inv, lvy = ly * inv, lvz = lz * inv;
        float avx = ax * inv, avy = ay * inv, avz = az * inv;
        // rotate back by q^-1
        float rlx, rly, rlz, rax, ray, raz;
        quat_rot(qw0, -qx, -qy, -qz, lvx, lvy, lvz, rlx, rly, rlz);
        quat_rot(qw0, -qx, -qy, -qz, avx, avy, avz, rax, ray, raz);
        // orbital = query_coord x lin_vel
        const float ccx = query_coord[q*3+0], ccy = query_coord[q*3+1], ccz = query_coord[q*3+2];
        const float obx = ccy * rlz - ccz * rly;
        const float oby = ccz * rlx - ccx * rlz;
        const float obz = ccx * rly - ccy * rlx;
        const float qwt = query_weight[q];
        red[wave][0] = qwt * (obx + rax * INV_ANG_MULT);
        red[wave][1] = qwt * (oby + ray * INV_ANG_MULT);
        red[wave][2] = qwt * (obz + raz * INV_ANG_MULT);
        red[wave][3] = qwt * rlx;   // LIN_MULT == 1
        red[wave][4] = qwt * rly;
        red[wave][5] = qwt * rlz;
    }
    __syncthreads();

    if (threadIdx.x < 6) {
        float s = 0.f;
#pragma unroll
        for (int w = 0; w < 8; ++w) s += red[w][threadIdx.x];
        partial[blockIdx.x * 6 + threadIdx.x] = s;
    }
}

// ---------------------------------------------------------------------------
// Kernel 4: final reduce. 32 blocks per t -> out. ang_out first, then lin_out.
// ---------------------------------------------------------------------------
__global__ void reduce_kernel(const float* __restrict__ partial,  // [1024,6]
                              float* __restrict__ out) {          // [192]
    const int tid = threadIdx.x;  // 192 threads
    if (tid >= N_T * 6) return;
    const int t = tid / 6, comp = tid % 6;
    float s = 0.f;
    for (int b = 0; b < 32; ++b) s += partial[(t * 32 + b) * 6 + comp];
    if (comp < 3) out[t * 3 + comp] = s;                 // ang_out [32,3]
    else          out[N_T * 3 + t * 3 + (comp - 3)] = s; // lin_out [32,3]
}

// ---------------------------------------------------------------------------
// Host launcher
// ---------------------------------------------------------------------------
extern "C" void kernel_launch(void* const* d_in, const int* in_sizes, int n_in,
                              void* d_out, int out_size, void* d_ws, size_t ws_size,
                              hipStream_t stream) {
    const float* T            = (const float*)d_in[0];
    const float* query_weight = (const float*)d_in[1];
    const float* query_feat   = (const float*)d_in[2];
    const float* query_coord  = (const float*)d_in[3];
    const float* node_feature = (const float*)d_in[4];
    const float* node_coord   = (const float*)d_in[5];
    const float* time_emb     = (const float*)d_in[6];
    const float* W_ext        = (const float*)d_in[7];
    const float* W_rbf        = (const float*)d_in[8];
    const float* W_time       = (const float*)d_in[9];
    const float* w_path_lin   = (const float*)d_in[10];
    // d_in[11] = W_scal_lin : dead code ([:,1:] drops the scalar column)
    const float* W_vec_lin    = (const float*)d_in[12];
    const float* w_path_ang   = (const float*)d_in[13];
    // d_in[14] = W_scal_ang : dead code
    const float* W_vec_ang    = (const float*)d_in[15];
    const int*   query_batch  = (const int*)d_in[16];
    const int*   node_batch   = (const int*)d_in[17];
    float* out = (float*)d_out;

    // workspace layout (all 16B-aligned chunks)
    float* ws       = (float*)d_ws;
    float* NF       = ws;                                   // 8192*128
    float* tw       = NF + (size_t)N_NODE * D_FEAT;         // 128
    float* coef     = tw + 128;                             // 192
    float* knn_d    = coef + 192;                           // 8192*16
    int*   knn_idx  = (int*)(knn_d + (size_t)N_PTS * K_NN); // 8192*16
    float* partial  = (float*)(knn_idx + (size_t)N_PTS * K_NN); // 1024*6

    precompute_kernel<<<1, 128, 0, stream>>>(time_emb, W_time,
                                             w_path_lin, W_vec_lin,
                                             w_path_ang, W_vec_ang,
                                             tw, coef);

    // 4096 16x16 tiles, 8 waves/block -> 512 blocks
    gemm_nf_kernel<<<512, 256, 0, stream>>>(node_feature, W_ext, NF);

    // 8192 query points, thread each
    knn_kernel<<<N_PTS / 256, 256, 0, stream>>>(T, query_coord, query_batch,
                                                node_coord, node_batch,
                                                knn_idx, knn_d);

    // 1024 blocks (32 per t), wave-per-query
    field_dtp_kernel<<<1024, 256, 0, stream>>>(T, query_weight, query_feat,
                                               query_coord, W_rbf, NF, tw, coef,
                                               knn_idx, knn_d, partial);

    reduce_kernel<<<1, 192, 0, stream>>>(partial, out);
}